// EdgeLoss_32006096290570
// MI455X (gfx1250) — compile-verified
//
#include <hip/hip_runtime.h>
#include <math.h>

// Problem shape from the reference setup_inputs(): [B,C,H,W] = [16,1,512,512]
#define B_   16
#define H_   512
#define W_   512
#define NROW (B_ * H_)         // 8192 rows
#define NCOL (B_ * W_)         // 8192 columns
#define NPIX (B_ * H_ * W_)    // 4,194,304 pixels
#define INF_ 1.0e6f            // sentinel distance (matches reference)
#define BIG_ 1.0e12f           // INF_^2

typedef __attribute__((ext_vector_type(2))) float v2f;
typedef __attribute__((ext_vector_type(8))) float v8f;

// ---------------------------------------------------------------------------
// Wave32 fp32 sum using the CDNA5 matrix core:
//   V_WMMA_F32_16X16X4_F32 with B = ones computes D[m,n] = sum_k A[m,k] + C.
// A layout (32-bit 16x4): lanes 0-15 supply rows m=0..15 {K0,K1}, lanes 16-31
// supply {K2,K3}. We place each lane's value in element 0 and zero in element
// 1, so D[m,n] = val[m] + val[m+16] for every n. Every lane then holds an
// identical column of D in its 8 accumulator VGPRs; summing them and folding
// lanes 0-15 with 16-31 via one shfl_xor yields the full wave sum in fp32
// with a fixed, deterministic reduction order.
// ---------------------------------------------------------------------------
__device__ __forceinline__ float wave_sum_wmma(float val) {
    v2f a;  a.x = val;  a.y = 0.0f;
    v2f bo; bo.x = 1.0f; bo.y = 1.0f;
    v8f c = {};
    c = __builtin_amdgcn_wmma_f32_16x16x4_f32(
        /*neg_a=*/false, a, /*neg_b=*/false, bo,
        /*c_mod=*/(short)0, c, /*reuse_a=*/false, /*reuse_b=*/false);
    float s = c[0] + c[1] + c[2] + c[3] + c[4] + c[5] + c[6] + c[7];
    s += __shfl_xor(s, 16, 32);    // fold M=0..7(+16..23) with M=8..15(+24..31)
    return s;                      // wave total, valid on all lanes
}

// ---------------------------------------------------------------------------
// Pass 1: per-column 1D scan along H for BOTH masks (tgt and 1-tgt) in one
// sweep. Forward scan stores the running distance; backward scan combines,
// clamps to INF and squares in place. One thread per (b, w) column; lanes map
// to consecutive w -> fully coalesced 128B/wave accesses.
// ---------------------------------------------------------------------------
__global__ void edt_cols(const int* __restrict__ tgt,
                         float* __restrict__ F0,   // squared vdist, mask = 1-tgt
                         float* __restrict__ F1) { // squared vdist, mask = tgt
    int tid = blockIdx.x * blockDim.x + threadIdx.x;   // 0 .. NCOL-1
    int b = tid >> 9;              // / W_
    int w = tid & (W_ - 1);
    int base = b * (H_ * W_) + w;

    float d0 = INF_, d1 = INF_;
    for (int h = 0; h < H_; ++h) {
        int idx = base + h * W_;
        int t = tgt[idx];
        d0 = (t == 0) ? d0 + 1.0f : 0.0f;   // foreground of (1 - tgt)
        d1 = (t != 0) ? d1 + 1.0f : 0.0f;   // foreground of tgt
        F0[idx] = d0;
        F1[idx] = d1;
    }
    d0 = INF_; d1 = INF_;
    for (int h = H_ - 1; h >= 0; --h) {
        int idx = base + h * W_;
        int t = tgt[idx];
        d0 = (t == 0) ? d0 + 1.0f : 0.0f;
        d1 = (t != 0) ? d1 + 1.0f : 0.0f;
        float g0 = fminf(fminf(F0[idx], d0), INF_);
        float g1 = fminf(fminf(F1[idx], d1), INF_);
        F0[idx] = g0 * g0;
        F1[idx] = g1 * g1;
    }
}

// ---------------------------------------------------------------------------
// Pass 2 + fused loss: one 256-thread block per row. Stage both f-rows in LDS
// (broadcast reads: every lane reads the same k -> no bank conflicts), brute
// force D[j] = min_k (j-k)^2 + f[k] for two j per thread, then sqrt/sigmoid/
// product, 1/N scaling, and a WMMA-based deterministic block reduction.
// ---------------------------------------------------------------------------
__global__ void edt_rows_loss(const float* __restrict__ logits,
                              const float* __restrict__ F0,
                              const float* __restrict__ F1,
                              float* __restrict__ P) {
    __shared__ float s0[W_];
    __shared__ float s1[W_];
    __shared__ float red[8];

    int row  = blockIdx.x;         // 0 .. NROW-1   (== b*H + h, since C==1)
    int tid  = threadIdx.x;        // 0 .. 255
    int base = row * W_;

    __builtin_prefetch(&logits[base + tid], 0, 0);   // global_prefetch_b8

    s0[tid]       = F0[base + tid];
    s0[tid + 256] = F0[base + tid + 256];
    s1[tid]       = F1[base + tid];
    s1[tid + 256] = F1[base + tid + 256];
    __syncthreads();

    int j  = tid;
    int j2 = tid + 256;
    float D0a = BIG_, D1a = BIG_, D0b = BIG_, D1b = BIG_;
    float da = (float)j;           // (j - k), maintained incrementally (exact:
    float db = (float)j2;          //  all values are small integers in fp32)
#pragma unroll 8
    for (int k = 0; k < W_; ++k) {
        float f0 = s0[k];
        float f1 = s1[k];
        float ca = da * da;
        float cb = db * db;
        D0a = fminf(D0a, ca + f0);
        D1a = fminf(D1a, ca + f1);
        D0b = fminf(D0b, cb + f0);
        D1b = fminf(D1b, cb + f1);
        da -= 1.0f;
        db -= 1.0f;
    }

    float xa = logits[base + j];
    float xb = logits[base + j2];
    float pa = 1.0f / (1.0f + expf(-xa));
    float pb = 1.0f / (1.0f + expf(-xb));
    float va = pa * (sqrtf(D0a) - sqrtf(D1a));   // edt(1-tgt) - edt(tgt)
    float vb = pb * (sqrtf(D0b) - sqrtf(D1b));
    float val = va + vb;

    float wsum = wave_sum_wmma(val);             // v_wmma_f32_16x16x4_f32
    if ((tid & 31) == 0) red[tid >> 5] = wsum;
    __syncthreads();
    if (tid == 0) {
        float t = 0.0f;
#pragma unroll
        for (int i = 0; i < 8; ++i) t += red[i];
        P[row] = t * (1.0f / (float)NPIX);
    }
}

// ---------------------------------------------------------------------------
// Final deterministic reduction of NROW partials -> scalar mean.
// ---------------------------------------------------------------------------
__global__ void reduce_final(const float* __restrict__ P,
                             float* __restrict__ outp) {
    __shared__ float red[8];
    int tid = threadIdx.x;
    float v = 0.0f;
    for (int i = tid; i < NROW; i += 256) v += P[i];
    float wsum = wave_sum_wmma(v);
    if ((tid & 31) == 0) red[tid >> 5] = wsum;
    __syncthreads();
    if (tid == 0) {
        float t = 0.0f;
#pragma unroll
        for (int i = 0; i < 8; ++i) t += red[i];
        outp[0] = t;
    }
}

extern "C" void kernel_launch(void* const* d_in, const int* in_sizes, int n_in,
                              void* d_out, int out_size, void* d_ws, size_t ws_size,
                              hipStream_t stream) {
    const float* logits = (const float*)d_in[0];   // "output" [16,1,512,512] f32
    const int*   tgt    = (const int*)d_in[1];     // "target" [16,1,512,512] i32

    // Workspace layout: F0 | F1 (NPIX floats each) | P (NROW floats)
    float* F0 = (float*)d_ws;
    float* F1 = F0 + NPIX;
    float* P  = F1 + NPIX;

    edt_cols<<<NCOL / 256, 256, 0, stream>>>(tgt, F0, F1);
    edt_rows_loss<<<NROW, 256, 0, stream>>>(logits, F0, F1, P);
    reduce_final<<<1, 256, 0, stream>>>(P, (float*)d_out);
}